// GateRecurrent2dnoind_11467562680435
// MI455X (gfx1250) — compile-verified
//
#include <hip/hip_runtime.h>
#include <stdint.h>

// -------- CDNA5 (gfx1250) SPN left-to-right scan, TDM-staged --------
// X,G1,G2,G3 : [N=8, C=64, H=256, W=256] f32 ; out same shape.
// One block per (n,c) plane; thread h owns row h; sequential over W.

typedef __attribute__((ext_vector_type(4))) unsigned int v4u;
typedef __attribute__((ext_vector_type(8))) int          v8i;
typedef __attribute__((ext_vector_type(4))) int          v4i;

#define HH 256
#define WW 256
#define WT 8            // columns per chunk (tile_dim0)
#define RS 9            // padded LDS row stride in dwords (8 data + 1 pad)
#define NCHUNK (WW / WT)

// Issue one TDM 2D tile load: [tile rows = HH] x [tile cols = WT] f32 elements,
// global row stride = WW elements, LDS rows padded to RS dwords.
__device__ __forceinline__ void tdm_load_tile(const float* gptr, uint64_t elem_off,
                                              uint32_t lds_byte_addr) {
  uint64_t ga = (uint64_t)(uintptr_t)gptr + elem_off * 4ull;

  v4u g0;
  g0[0] = 1u;                                   // count=1 (valid user descriptor)
  g0[1] = lds_byte_addr;                        // LDS destination (bytes)
  g0[2] = (uint32_t)ga;                         // global_addr[31:0]
  g0[3] = (uint32_t)(ga >> 32) | (2u << 30);    // global_addr[56:32] | type=2 (image)

  v8i g1;
  // data_size=2 (4B) | pad_enable | pad_interval=2 (every 8 dwords) | pad_amount=0 (1 dword)
  g1[0] = (int)((2u << 16) | (1u << 20) | (2u << 22));
  g1[1] = (int)(((uint32_t)WW & 0xFFFFu) << 16);            // tensor_dim0[15:0]
  g1[2] = (int)((((uint32_t)HH & 0xFFFFu) << 16) |          // tensor_dim1[15:0]
                (((uint32_t)WW >> 16) & 0xFFFFu));          // tensor_dim0[31:16]
  g1[3] = (int)((((uint32_t)WT) << 16) |                    // tile_dim0 = 8
                (((uint32_t)HH >> 16) & 0xFFFFu));          // tensor_dim1[31:16]
  g1[4] = (int)HH;                                          // tile_dim1 = 256 (tile_dim2 = 0)
  g1[5] = (int)WW;                                          // tensor_dim0_stride[31:0] = 256
  g1[6] = (int)((((uint32_t)(HH * WW)) & 0xFFFFu) << 16);   // tensor_dim1_stride[15:0]
  g1[7] = (int)(((uint32_t)(HH * WW)) >> 16);               // tensor_dim1_stride[47:16]

  v4i g2 = {0, 0, 0, 0};                        // 2D tensor: dims 2/3 unused
  v4i g3 = {0, 0, 0, 0};
  v8i g4 = {0, 0, 0, 0, 0, 0, 0, 0};            // 6-arg toolchain: extra group (unused)

  __builtin_amdgcn_tensor_load_to_lds(g0, g1, g2, g3, g4, /*cpol*/0);
}

__global__ __launch_bounds__(256)
void spn_scan_kernel(const float* __restrict__ X,  const float* __restrict__ G1,
                     const float* __restrict__ G2, const float* __restrict__ G3,
                     float* __restrict__ out) {
  // Double-buffered staged tiles: [buf][tensor][HH rows * RS dwords]
  __shared__ float tile[2][4][HH * RS];
  // Double-buffered h column with zero halo cells [0] and [HH+1]
  __shared__ float hcol[2][HH + 2];

  const int h = threadIdx.x;                    // row owned by this thread
  const uint64_t pbase = (uint64_t)blockIdx.x * (uint64_t)(HH * WW);

  // h_prev initial state = 0; halo cells stay 0 forever (writes go to [1..HH]).
  for (int i = h; i < 2 * (HH + 2); i += blockDim.x)
    ((float*)hcol)[i] = 0.0f;

  const bool loader = (h < 32);                 // wave 0 drives the TDM

  if (loader) {                                 // prologue: chunk 0 -> buffer 0
    tdm_load_tile(X,  pbase, (uint32_t)(size_t)&tile[0][0][0]);
    tdm_load_tile(G1, pbase, (uint32_t)(size_t)&tile[0][1][0]);
    tdm_load_tile(G2, pbase, (uint32_t)(size_t)&tile[0][2][0]);
    tdm_load_tile(G3, pbase, (uint32_t)(size_t)&tile[0][3][0]);
  }

  int cur = 0;
  float acc[4];

  for (int c = 0; c < NCHUNK; ++c) {
    const int buf = c & 1;

    if (loader) __builtin_amdgcn_s_wait_tensorcnt(0);
    __syncthreads();   // tile[buf] ready for all waves; compute on tile[buf^1] finished

    if (loader && (c + 1) < NCHUNK) {           // prefetch next chunk into other buffer
      const uint64_t off = pbase + (uint64_t)(c + 1) * WT;
      tdm_load_tile(X,  off, (uint32_t)(size_t)&tile[buf ^ 1][0][0]);
      tdm_load_tile(G1, off, (uint32_t)(size_t)&tile[buf ^ 1][1][0]);
      tdm_load_tile(G2, off, (uint32_t)(size_t)&tile[buf ^ 1][2][0]);
      tdm_load_tile(G3, off, (uint32_t)(size_t)&tile[buf ^ 1][3][0]);
    }

    const float* tx  = &tile[buf][0][h * RS];   // bank-conflict-free: gcd(RS,64)=1
    const float* tg1 = &tile[buf][1][h * RS];
    const float* tg2 = &tile[buf][2][h * RS];
    const float* tg3 = &tile[buf][3][h * RS];

#pragma unroll
    for (int wi = 0; wi < WT; ++wi) {
      const float xv = tx[wi];
      const float a1 = tg1[wi];
      const float a2 = tg2[wi];
      const float a3 = tg3[wi];

      const float* hc = hcol[cur];
      float hv = xv * (1.0f - a1 - a2 - a3);
      hv = fmaf(a1, hc[h],     hv);             // h_prev[h-1]  (halo -> 0 at h==0)
      hv = fmaf(a2, hc[h + 1], hv);             // h_prev[h]
      hv = fmaf(a3, hc[h + 2], hv);             // h_prev[h+1]  (halo -> 0 at h==HH-1)

      hcol[cur ^ 1][h + 1] = hv;                // publish for next column
      acc[wi & 3] = hv;
      if ((wi & 3) == 3) {                      // per-lane b128 store
        float4 v = make_float4(acc[0], acc[1], acc[2], acc[3]);
        *(float4*)(out + pbase + (uint64_t)h * WW + (uint64_t)(c * WT + wi - 3)) = v;
      }

      cur ^= 1;
      __syncthreads();                          // one barrier per W-step
    }
  }
}

extern "C" void kernel_launch(void* const* d_in, const int* in_sizes, int n_in,
                              void* d_out, int out_size, void* d_ws, size_t ws_size,
                              hipStream_t stream) {
  (void)in_sizes; (void)n_in; (void)out_size; (void)d_ws; (void)ws_size;
  const float* X  = (const float*)d_in[0];
  const float* G1 = (const float*)d_in[1];
  const float* G2 = (const float*)d_in[2];
  const float* G3 = (const float*)d_in[3];
  float* out = (float*)d_out;

  dim3 grid(8 * 64);    // one block per (n,c) plane
  dim3 block(256);      // 8 wave32s; thread h owns row h
  spn_scan_kernel<<<grid, block, 0, stream>>>(X, G1, G2, G3, out);
}